// TactileFeatureEncoder_87170656240171
// MI455X (gfx1250) — compile-verified
//
#include <hip/hip_runtime.h>
#include <hip/hip_bf16.h>

typedef float v2f __attribute__((ext_vector_type(2)));
typedef float v8f __attribute__((ext_vector_type(8)));

#define BN_EPS_F 1e-5f

// ---------------- LDS layout (float offsets) ----------------
// LX  : zero-padded sample  [11][34][34]  (C pad 1 each side, H/W pad 1)
// U   : union region: t1 = S_W(x) [11][34][34] (12716)  -> conv1 relu out [16][32][32] (16384)
// POOL: maxpooled [16][16][16]
#define OFF_LX    0
#define LX_SZ     12716      // 11*34*34
#define OFF_U     12720
#define U_SZ      16384      // 16*1024
#define OFF_POOL  29104      // 4096
#define OFF_W1    33200      // 576
#define OFF_S1    33776      // 16  (bn1 scale)
#define OFF_SH1   33792      // 16  (bn1 shift, conv bias folded)
#define OFF_W2    33808      // 2048
#define OFF_S2    35856      // 32
#define OFF_SH2   35888      // 32
#define OFF_FW1   35920      // 2048
#define OFF_FB1   37968      // 64
#define OFF_FW2   38032      // 1024
#define OFF_FB2   39056      // 16
#define OFF_RED   39072      // 8 waves * 12 partials
#define OFF_H1    39200      // 64
#define OFF_GAP   39264      // 32
#define LDS_FLOATS 39296     // 157184 bytes

__device__ __forceinline__ float wsum(float v){
#pragma unroll
  for (int o = 16; o; o >>= 1) v += __shfl_xor(v, o, 32);
  return v;
}
__device__ __forceinline__ float wmaxr(float v){
#pragma unroll
  for (int o = 16; o; o >>= 1) v = fmaxf(v, __shfl_xor(v, o, 32));
  return v;
}
__device__ __forceinline__ float wminr(float v){
#pragma unroll
  for (int o = 16; o; o >>= 1) v = fminf(v, __shfl_xor(v, o, 32));
  return v;
}

__global__ __launch_bounds__(256)
void tactile_encoder_kernel(
    const float* __restrict__ x,
    const float* __restrict__ c1w, const float* __restrict__ c1b,
    const float* __restrict__ g1,  const float* __restrict__ be1,
    const float* __restrict__ mu1, const float* __restrict__ va1,
    const float* __restrict__ c2w, const float* __restrict__ c2b,
    const float* __restrict__ g2,  const float* __restrict__ be2,
    const float* __restrict__ mu2, const float* __restrict__ va2,
    const float* __restrict__ f1w, const float* __restrict__ f1b,
    const float* __restrict__ f2w, const float* __restrict__ f2b,
    float* __restrict__ out)
{
  extern __shared__ float sm[];
  const int tid  = threadIdx.x;
  const int b    = blockIdx.x;
  const int lane = tid & 31;
  const int wid  = tid >> 5;
  // wave id as a compiler-visible scalar -> uniform SGPR tile loops (EXEC stays all-1s
  // around every WMMA without exec-masked loop structure)
  const int swid = __builtin_amdgcn_readfirstlane(wid);

  // ---------- Phase 0: zero padded regions, stage weights, fold BN ----------
  for (int i = tid; i < LX_SZ; i += 256) sm[OFF_LX + i] = 0.f;
  for (int i = tid; i < LX_SZ; i += 256) sm[OFF_U  + i] = 0.f;   // t1 borders
  if (tid < 32) sm[OFF_GAP + tid] = 0.f;
  for (int i = tid; i < 576;  i += 256) sm[OFF_W1  + i] = c1w[i];
  for (int i = tid; i < 2048; i += 256) sm[OFF_W2  + i] = c2w[i];
  for (int i = tid; i < 2048; i += 256) sm[OFF_FW1 + i] = f1w[i];
  for (int i = tid; i < 1024; i += 256) sm[OFF_FW2 + i] = f2w[i];
  if (tid < 64) sm[OFF_FB1 + tid] = f1b[tid];
  if (tid < 16) sm[OFF_FB2 + tid] = f2b[tid];
  if (tid < 16) {
    float s = g1[tid] * rsqrtf(va1[tid] + BN_EPS_F);
    sm[OFF_S1  + tid] = s;
    sm[OFF_SH1 + tid] = be1[tid] + (c1b[tid] - mu1[tid]) * s;
  }
  if (tid < 32) {
    float s = g2[tid] * rsqrtf(va2[tid] + BN_EPS_F);
    sm[OFF_S2  + tid] = s;
    sm[OFF_SH2 + tid] = be2[tid] + (c2b[tid] - mu2[tid]) * s;
  }
  __syncthreads();

  // ---------- Phase 1: stage x[b] into padded LDS with async global->LDS DMA ----------
  // GLOBAL_LOAD_ASYNC_TO_LDS_B32 (GVS mode): per-lane global byte offset + per-lane LDS
  // byte address -> does the padded scatter with no VGPR round trip. No static __shared__
  // in this kernel, so the dynamic-LDS base is 0 and the LDS address is just idx*4.
  {
    const unsigned long long gbase = (unsigned long long)(x + (size_t)b * 9216);
    for (int e = tid; e < 9216; e += 256) {
      int c = e >> 10, h = (e >> 5) & 31, w = e & 31;
      unsigned lds  = (unsigned)((OFF_LX + (c + 1) * 1156 + (h + 1) * 34 + (w + 1)) << 2);
      unsigned goff = (unsigned)(e << 2);
      asm volatile("global_load_async_to_lds_b32 %0, %1, %2"
                   :: "v"(lds), "v"(goff), "s"(gbase) : "memory");
    }
    asm volatile("s_wait_asynccnt 0x0" ::: "memory");
  }
  __syncthreads();

  // ---------- Phase 2: t1 = [1,2,1]-smooth along W (shared by both sobels) ----------
  for (int e = tid; e < 9216; e += 256) {
    int c = e >> 10, h = (e >> 5) & 31, w = e & 31;
    int base = OFF_LX + (c + 1) * 1156 + (h + 1) * 34 + w;   // w-1 in padded coords
    sm[OFF_U + (c + 1) * 1156 + (h + 1) * 34 + (w + 1)] =
        sm[base] + 2.f * sm[base + 1] + sm[base + 2];
  }
  __syncthreads();

  // ---------- Phase 3: fused manual features ----------
  float p_sum = 0.f, p_sq = 0.f, p_max = -__builtin_inff(), p_min = __builtin_inff();
  float p_g = 0.f, p_cx = 0.f, p_cy = 0.f, p_hs = 0.f, p_vs = 0.f;
  float p_es = 0.f, p_eq = 0.f, p_em = -__builtin_inff();
  for (int e = tid; e < 9216; e += 256) {
    int c = e >> 10, h = (e >> 5) & 31, w = e & 31;
    int li = OFF_LX + (c + 1) * 1156 + (h + 1) * 34 + (w + 1);
    float v = sm[li];
    p_sum += v; p_sq += v * v;
    p_max = fmaxf(p_max, v); p_min = fminf(p_min, v);
    // np.gradient along C (axis 0 of [C,H,W]) and H (axis 1): one-sided edges
    float gx = (c == 0) ? sm[li + 1156] - v
             : (c == 8) ? v - sm[li - 1156]
             : 0.5f * (sm[li + 1156] - sm[li - 1156]);
    float gy = (h == 0)  ? sm[li + 34] - v
             : (h == 31) ? v - sm[li - 34]
             : 0.5f * (sm[li + 34] - sm[li - 34]);
    p_g  += sqrtf(gx * gx + gy * gy);
    p_cx += v * (float)w;
    p_cy += v * (float)h;
    p_hs += fabsf(v - sm[OFF_LX + (c + 1) * 1156 + (32 - h) * 34 + (w + 1)]);  // flip H
    p_vs += fabsf(v - sm[OFF_LX + (9 - c) * 1156 + (h + 1) * 34 + (w + 1)]);  // flip C
    // sobel: sx = D_C(S_H(t1)), sy = S_C(D_H(t1)); zero padding via LDS borders
    int tb = OFF_U + c * 1156 + h * 34 + (w + 1);
    float a0 = sm[tb]        + 2.f * sm[tb + 34]        + sm[tb + 68];
    float b0 = sm[tb + 68]   -       sm[tb];
    float a2 = sm[tb + 2312] + 2.f * sm[tb + 2312 + 34] + sm[tb + 2312 + 68];
    float b1 = sm[tb + 1156 + 68] - sm[tb + 1156];
    float b2 = sm[tb + 2312 + 68] - sm[tb + 2312];
    float sx = a2 - a0;
    float sy = b0 + 2.f * b1 + b2;
    float em = sqrtf(sx * sx + sy * sy);
    p_es += em; p_eq += em * em; p_em = fmaxf(p_em, em);
  }
  p_sum = wsum(p_sum); p_sq = wsum(p_sq); p_max = wmaxr(p_max); p_min = wminr(p_min);
  p_g = wsum(p_g); p_cx = wsum(p_cx); p_cy = wsum(p_cy); p_hs = wsum(p_hs);
  p_vs = wsum(p_vs); p_es = wsum(p_es); p_eq = wsum(p_eq); p_em = wmaxr(p_em);
  if (lane == 0) {
    float* R = &sm[OFF_RED + wid * 12];
    R[0] = p_sum; R[1] = p_sq; R[2] = p_max; R[3] = p_min; R[4] = p_g; R[5] = p_cx;
    R[6] = p_cy; R[7] = p_hs; R[8] = p_vs; R[9] = p_es; R[10] = p_eq; R[11] = p_em;
  }
  __syncthreads();   // also fences phase-3 reads of t1 before conv1 overwrites U

  if (tid == 0) {
    float F[12];
#pragma unroll
    for (int f = 0; f < 12; ++f) {
      float a;
      if (f == 2 || f == 11) { a = -__builtin_inff(); for (int w8 = 0; w8 < 8; ++w8) a = fmaxf(a, sm[OFF_RED + w8 * 12 + f]); }
      else if (f == 3)       { a =  __builtin_inff(); for (int w8 = 0; w8 < 8; ++w8) a = fminf(a, sm[OFF_RED + w8 * 12 + f]); }
      else                   { a = 0.f;               for (int w8 = 0; w8 < 8; ++w8) a += sm[OFF_RED + w8 * 12 + f]; }
      F[f] = a;
    }
    const float N = 9216.f;
    float mean = F[0] / N;
    float stdv = sqrtf(fmaxf(F[1] / N - mean * mean, 0.f));
    float tot  = F[0];
    float copx = (tot > 0.f) ? (F[5] / tot) * (1.f / 32.f) : 0.5f;
    float copy = (tot > 0.f) ? (F[6] / tot) * (1.f / 32.f) : 0.5f;
    float sym  = 1.f / (1.f + F[7] / N + F[8] / N);
    float emm  = F[9] / N;
    float emsd = sqrtf(fmaxf(F[10] / N - emm * emm, 0.f));
    float* o = out + (size_t)b * 27 + 16;
    o[0] = mean; o[1] = stdv; o[2] = F[2]; o[3] = F[3]; o[4] = F[4] / N;
    o[5] = copx; o[6] = copy; o[7] = sym; o[8] = emm; o[9] = emsd; o[10] = F[11];
  }

  // ---------- Phase 4: conv1 via V_WMMA_F32_16X16X4_F32 (implicit GEMM) ----------
  // M = 1024 positions (rows/cols 0..31 of the padded 33x33 map; row/col 32 never pooled)
  // N = 16 oc, K = 36 (ic*4+tap) = 9 steps of K=4; step s == input channel s.
  const int koff = (lane >> 4) << 1;    // lanes 0-15: K 0,1 ; lanes 16-31: K 2,3
  const int drL  = lane >> 4;           // tap row for this half-wave
  const int nN   = lane & 15;
  float bA1[9], bB1[9];
#pragma unroll
  for (int s = 0; s < 9; ++s) {
    int bi = OFF_W1 + nN * 36 + s * 4 + koff;
    bA1[s] = sm[bi]; bB1[s] = sm[bi + 1];
  }
  for (int t = swid; t < 64; t += 8) {
    int p = t * 16 + (lane & 15);
    int i = p >> 5, col = p & 31;
    v8f acc = {0.f, 0.f, 0.f, 0.f, 0.f, 0.f, 0.f, 0.f};
#pragma unroll
    for (int s = 0; s < 9; ++s) {
      int ai = OFF_LX + (s + 1) * 1156 + (i + drL) * 34 + col;
      v2f A; A[0] = sm[ai]; A[1] = sm[ai + 1];
      v2f Bv; Bv[0] = bA1[s]; Bv[1] = bB1[s];
      acc = __builtin_amdgcn_wmma_f32_16x16x4_f32(false, A, false, Bv,
                                                  (short)0, acc, false, false);
    }
    float sc = sm[OFF_S1 + nN], sh = sm[OFF_SH1 + nN];
    int mbase = drL * 8;
#pragma unroll
    for (int r = 0; r < 8; ++r) {
      int pp = t * 16 + mbase + r;
      sm[OFF_U + nN * 1024 + pp] = fmaxf(acc[r] * sc + sh, 0.f);  // bias+BN+ReLU
    }
  }
  __syncthreads();

  // ---------- Phase 5: 2x2/2 maxpool -> POOL[16][16][16] ----------
  for (int e = tid; e < 4096; e += 256) {
    int oc = e >> 8, rem = e & 255, pr = rem >> 4, pc = rem & 15;
    int base = OFF_U + oc * 1024 + (pr * 2) * 32 + pc * 2;
    sm[OFF_POOL + e] = fmaxf(fmaxf(sm[base], sm[base + 1]),
                             fmaxf(sm[base + 32], sm[base + 33]));
  }
  __syncthreads();

  // ---------- Phase 6: conv2 via WMMA, fused BN+ReLU+global-avg-pool ----------
  // M = 225 positions -> 15 tiles (tail masked); N = 32 oc -> 2 N-tiles; K = 64 -> 16 steps.
  for (int t = swid; t < 15; t += 8) {
    int p = t * 16 + (lane & 15);
    if (p > 224) p = 224;                 // clamp A loads; outputs masked below
    int r = p / 15, c = p - r * 15;
    v8f a0 = {0.f, 0.f, 0.f, 0.f, 0.f, 0.f, 0.f, 0.f};
    v8f a1 = {0.f, 0.f, 0.f, 0.f, 0.f, 0.f, 0.f, 0.f};
#pragma unroll
    for (int s = 0; s < 16; ++s) {
      int ai = OFF_POOL + s * 256 + (r + drL) * 16 + c;
      v2f A; A[0] = sm[ai]; A[1] = sm[ai + 1];
      int b0i = OFF_W2 + nN * 64 + s * 4 + koff;
      int b1i = OFF_W2 + (nN + 16) * 64 + s * 4 + koff;
      v2f B0; B0[0] = sm[b0i]; B0[1] = sm[b0i + 1];
      v2f B1; B1[0] = sm[b1i]; B1[1] = sm[b1i + 1];
      a0 = __builtin_amdgcn_wmma_f32_16x16x4_f32(false, A, false, B0,
                                                 (short)0, a0, false, false);
      a1 = __builtin_amdgcn_wmma_f32_16x16x4_f32(false, A, false, B1,
                                                 (short)0, a1, false, false);
    }
    float sc0 = sm[OFF_S2 + nN],      sh0 = sm[OFF_SH2 + nN];
    float sc1 = sm[OFF_S2 + nN + 16], sh1 = sm[OFF_SH2 + nN + 16];
    float q0 = 0.f, q1 = 0.f;
    int mbase = drL * 8;
#pragma unroll
    for (int r8 = 0; r8 < 8; ++r8) {
      int pp = t * 16 + mbase + r8;
      if (pp < 225) {
        q0 += fmaxf(a0[r8] * sc0 + sh0, 0.f);
        q1 += fmaxf(a1[r8] * sc1 + sh1, 0.f);
      }
    }
    atomicAdd(&sm[OFF_GAP + nN],      q0);   // ds_add_f32
    atomicAdd(&sm[OFF_GAP + nN + 16], q1);
  }
  __syncthreads();

  // ---------- Phase 7: FC1(relu) + FC2 -> enc[16] ----------
  if (tid < 64) {
    float acc = 0.f;
#pragma unroll
    for (int k = 0; k < 32; ++k) acc += sm[OFF_FW1 + tid * 32 + k] * sm[OFF_GAP + k];
    sm[OFF_H1 + tid] = fmaxf(sm[OFF_FB1 + tid] + acc * (1.f / 225.f), 0.f);
  }
  __syncthreads();
  if (tid < 16) {
    float acc = sm[OFF_FB2 + tid];
#pragma unroll
    for (int j = 0; j < 64; ++j) acc += sm[OFF_FW2 + tid * 64 + j] * sm[OFF_H1 + j];
    out[(size_t)b * 27 + tid] = acc;
  }
}

extern "C" void kernel_launch(void* const* d_in, const int* in_sizes, int n_in,
                              void* d_out, int out_size, void* d_ws, size_t ws_size,
                              hipStream_t stream) {
  (void)n_in; (void)out_size; (void)d_ws; (void)ws_size;
  const float* x   = (const float*)d_in[0];
  const float* c1w = (const float*)d_in[1];
  const float* c1b = (const float*)d_in[2];
  const float* g1  = (const float*)d_in[3];
  const float* be1 = (const float*)d_in[4];
  const float* mu1 = (const float*)d_in[5];
  const float* va1 = (const float*)d_in[6];
  const float* c2w = (const float*)d_in[7];
  const float* c2b = (const float*)d_in[8];
  const float* g2  = (const float*)d_in[9];
  const float* be2 = (const float*)d_in[10];
  const float* mu2 = (const float*)d_in[11];
  const float* va2 = (const float*)d_in[12];
  const float* f1w = (const float*)d_in[13];
  const float* f1b = (const float*)d_in[14];
  const float* f2w = (const float*)d_in[15];
  const float* f2b = (const float*)d_in[16];
  float* out = (float*)d_out;

  const int B = in_sizes[0] / (9 * 32 * 32);
  const size_t lds_bytes = (size_t)LDS_FLOATS * sizeof(float);
  (void)hipFuncSetAttribute((const void*)tactile_encoder_kernel,
                            hipFuncAttributeMaxDynamicSharedMemorySize,
                            (int)lds_bytes);
  tactile_encoder_kernel<<<B, 256, lds_bytes, stream>>>(
      x, c1w, c1b, g1, be1, mu1, va1,
      c2w, c2b, g2, be2, mu2, va2,
      f1w, f1b, f2w, f2b, out);
}